// ChannelWiseDense_29360396435731
// MI455X (gfx1250) — compile-verified
//
#include <hip/hip_runtime.h>
#include <hip/hip_bf16.h>

// B=32, H=W=32, C=128, N=H*W=1024
// y[b,c,m] = relu( sum_n x[b,n,c] * W[c,n,m] + bias[c,m] )
// out[b,h,w,cc] = y[b, 127-cc, h*32+w]

typedef float v2f __attribute__((ext_vector_type(2)));
typedef float v8f __attribute__((ext_vector_type(8)));

#define NB 32        // batch (GEMM M)
#define NN 1024      // spatial (GEMM K and output N)
#define NC 128       // channels

// ---------------------------------------------------------------------------
// Kernel 1: transpose x [B, N, C] -> AT [C, N, B]  (coalesced both directions
// via an LDS tile; one block per n, 256 threads)
// ---------------------------------------------------------------------------
__global__ __launch_bounds__(256) void cwd_transpose_kernel(
    const float* __restrict__ x, float* __restrict__ xt) {
  __shared__ float tile[NB][NC + 1];   // +1 pad: conflict-free on both phases
  const int n = blockIdx.x;            // 0..1023
  const int tid = threadIdx.x;

  // load 32(b) x 128(c): consecutive tid -> consecutive c (coalesced reads)
#pragma unroll
  for (int i = 0; i < 16; ++i) {
    int e = i * 256 + tid;             // 0..4095
    int b = e >> 7;                    // /128
    int c = e & 127;
    tile[b][c] = x[((long)b * NN + n) * NC + c];
  }
  __syncthreads();

  // store 128(c) x 32(b): consecutive tid -> consecutive b (coalesced writes)
#pragma unroll
  for (int i = 0; i < 16; ++i) {
    int e = i * 256 + tid;
    int c = e >> 5;                    // /32
    int b = e & 31;
    xt[(long)c * (NN * NB) + (long)n * NB + b] = tile[b][c];
  }
}

// ---------------------------------------------------------------------------
// Kernel 2: per-channel GEMM with V_WMMA_F32_16X16X4_F32.
// Grid: 128 channels * 8 column-blocks. Block: 128 threads = 4 waves.
// Wave w owns columns [colblk*128 + w*32, +32) and the full M=32, so each
// W element is fetched from HBM exactly once (non-temporal).
// A element (row=b, k=n) = Abase[c*aChan + b*aRow + n*aK]  (ws or direct x).
// ---------------------------------------------------------------------------
__global__ __launch_bounds__(128) void cwd_gemm_kernel(
    const float* __restrict__ Abase, long aChan, long aRow, long aK,
    const float* __restrict__ Wt, const float* __restrict__ bias,
    float* __restrict__ out) {
  const int wid  = threadIdx.x >> 5;
  const int lane = threadIdx.x & 31;
  const int half = lane >> 4;          // 0: lanes 0-15, 1: lanes 16-31
  const int lm   = lane & 15;

  const int c      = blockIdx.x >> 3;  // channel
  const int colblk = blockIdx.x & 7;
  const int n0     = colblk * 128 + wid * 32;   // first output column of wave

  const float* __restrict__ Ac = Abase + (long)c * aChan;
  const float* __restrict__ Wc = Wt + (long)c * ((long)NN * NN);

  v8f acc00 = {}, acc01 = {}, acc10 = {}, acc11 = {};

  // WMMA f32 A 16x4 layout: VGPR0 = K,K+2 (lane halves); VGPR1 = K+1,K+3.
  const long offA0 = (long)lm * aRow + (long)(half * 2) * aK;          // M-tile 0
  const long offA1 = (long)(16 + lm) * aRow + (long)(half * 2) * aK;   // M-tile 1
  // B 4x16 layout mirrors A: VGPR0 = rows K,K+2; VGPR1 = rows K+1,K+3.
  const long offB0 = (long)(half * 2) * NN + n0 + lm;                  // N-tile 0
  const long offB1 = offB0 + 16;                                      // N-tile 1

#pragma unroll 2
  for (int k = 0; k < NN; k += 4) {
    const long ka = (long)k * aK;
    v2f a0, a1, b0, b1;
    a0.x = Ac[offA0 + ka];
    a0.y = Ac[offA0 + ka + aK];
    a1.x = Ac[offA1 + ka];
    a1.y = Ac[offA1 + ka + aK];

    const long kb = (long)k * NN;
    b0.x = __builtin_nontemporal_load(&Wc[offB0 + kb]);
    b0.y = __builtin_nontemporal_load(&Wc[offB0 + kb + NN]);
    b1.x = __builtin_nontemporal_load(&Wc[offB1 + kb]);
    b1.y = __builtin_nontemporal_load(&Wc[offB1 + kb + NN]);

    acc00 = __builtin_amdgcn_wmma_f32_16x16x4_f32(false, a0, false, b0,
                                                  (short)0, acc00, false, false);
    acc01 = __builtin_amdgcn_wmma_f32_16x16x4_f32(false, a0, false, b1,
                                                  (short)0, acc01, false, false);
    acc10 = __builtin_amdgcn_wmma_f32_16x16x4_f32(false, a1, false, b0,
                                                  (short)0, acc10, false, false);
    acc11 = __builtin_amdgcn_wmma_f32_16x16x4_f32(false, a1, false, b1,
                                                  (short)0, acc11, false, false);
  }

  // Epilogue: C/D layout: VGPR r -> M = r + 8*half, N = lm (per tile).
  const int m0 = n0 + lm;          // output column, N-tile 0
  const int m1 = m0 + 16;          // output column, N-tile 1
  const float bv0 = bias[(long)c * NN + m0];
  const float bv1 = bias[(long)c * NN + m1];
  const int cc = (NC - 1) - c;     // reversed channel position

#pragma unroll
  for (int r = 0; r < 8; ++r) {
    const int b0r = r + half * 8;        // batch row, M-tile 0
    const int b1r = b0r + 16;            // batch row, M-tile 1
    out[(long)b0r * (NN * NC) + (long)m0 * NC + cc] = fmaxf(acc00[r] + bv0, 0.f);
    out[(long)b0r * (NN * NC) + (long)m1 * NC + cc] = fmaxf(acc01[r] + bv1, 0.f);
    out[(long)b1r * (NN * NC) + (long)m0 * NC + cc] = fmaxf(acc10[r] + bv0, 0.f);
    out[(long)b1r * (NN * NC) + (long)m1 * NC + cc] = fmaxf(acc11[r] + bv1, 0.f);
  }
}

extern "C" void kernel_launch(void* const* d_in, const int* in_sizes, int n_in,
                              void* d_out, int out_size, void* d_ws, size_t ws_size,
                              hipStream_t stream) {
  const float* x    = (const float*)d_in[0];   // [B, H, W, C] f32
  const float* Wt   = (const float*)d_in[1];   // [C, N, N]    f32
  const float* bias = (const float*)d_in[2];   // [C, N]       f32
  float* out = (float*)d_out;                  // [B, H, W, C] f32

  const size_t ws_needed = (size_t)NC * NN * NB * sizeof(float);  // 16 MB

  if (d_ws != nullptr && ws_size >= ws_needed) {
    // Phase 1: x [B,N,C] -> AT [C,N,B] so A-fragment loads are contiguous.
    cwd_transpose_kernel<<<dim3(NN), dim3(256), 0, stream>>>(x, (float*)d_ws);
    // Phase 2: GEMMs reading AT: Ac[b*1 + n*32], channel stride 32768.
    cwd_gemm_kernel<<<dim3(NC * 8), dim3(128), 0, stream>>>(
        (const float*)d_ws, (long)(NN * NB), 1L, (long)NB, Wt, bias, out);
  } else {
    // Fallback: read A directly from x (strided): x[b*131072 + n*128 + c].
    cwd_gemm_kernel<<<dim3(NC * 8), dim3(128), 0, stream>>>(
        x, 1L, (long)(NN * NC), (long)NC, Wt, bias, out);
  }
}